// old_VQ_19189913878562
// MI455X (gfx1250) — compile-verified
//
#include <hip/hip_runtime.h>
#include <hip/hip_bf16.h>
#include <math.h>

// ---------------------------------------------------------------------------
// VQ-VAE vector quantization for MI455X (gfx1250, wave32).
//   z:   (16,64,64,64) f32   -> z_flat (N=65536, D=64), n in BHW order
//   emb: (1024,64) f32       -> K=1024 codes
// Outputs flat in d_out (f32):
//   [0)            z_q_st        4194304
//   [4194304)      perplexity    1
//   [4194305)      indices       65536
//   [4259841)      min_encodings 67108864   (268MB -> NT stores dominate time)
//   [71368705)     loss          1
// ---------------------------------------------------------------------------

typedef float v2f __attribute__((ext_vector_type(2)));
typedef float v4f __attribute__((ext_vector_type(4)));
typedef float v8f __attribute__((ext_vector_type(8)));

#define NPTS   65536      // B*H*W
#define DCH    64
#define KCODE  1024
#define HW     4096       // H*W
#define NTOT   4194304    // B*D*H*W
#define NTILES 4096       // NPTS / 16
#define GRID_ARGMIN 128   // 128 WGs * 8 waves * 4 iters = 4096 tiles
#define CHUNK  128        // codes staged in LDS per pass (32KB, safe)

// ---------------------------------------------------------------------------
// Kernel 1: zero the code-usage histogram (deterministic re-init every call).
__global__ __launch_bounds__(1024) void vq_init(unsigned* hist) {
    hist[threadIdx.x] = 0u;
}

// ---------------------------------------------------------------------------
// Kernel 2: WMMA distance + argmin.
// Each wave owns a 16-row tile of z_flat. Codebook is staged in LDS in
// 128-code chunks as (-2*e), XOR-swizzled so the stride-64-word column reads
// hit distinct banks. Accumulator is seeded with ||e_k||^2, so after the
// K-dim chain of v_wmma_f32_16x16x4_f32 each C slot holds
// ||e_k||^2 - 2 z.e_k  (row-constant ||z||^2 dropped: argmin-invariant).
__global__ __launch_bounds__(256) void vq_argmin(
    const float* __restrict__ z, const float* __restrict__ emb,
    int* __restrict__ wsIdx, float* __restrict__ outIdxF,
    unsigned* __restrict__ hist)
{
    __shared__ float ldsB[CHUNK * DCH];  // -2*emb chunk, xor-swizzled
    __shared__ float ldsE[CHUNK];        // ||e||^2 chunk

    const int tid  = threadIdx.x;
    const int lane = tid & 31;
    const int wv   = tid >> 5;       // 8 waves per WG
    const int col  = lane & 15;      // C/D column (N) for this lane
    const int lh   = lane >> 4;      // lane half: selects K sub-pair / M+8 rows

    for (int it = 0; it < 4; ++it) {
        const int t = (it * GRID_ARGMIN + blockIdx.x) * 8 + wv;  // tile id

        // ---- Load A tile (16 rows x 64 ch) once, per documented A layout:
        // lanes 0-15: M=lane, VGPR0/1 = K pair (2*lh .. 2*lh+1) per 4-step.
        const int t16 = t << 4;
        const int bI  = t16 >> 12;                 // batch
        const int hw  = (t16 & 4095) + col;        // h*64 + w (contiguous w)
        const float* zp = z + (size_t)bI * (DCH * HW) + hw;
        v2f aReg[16];
#pragma unroll
        for (int s = 0; s < 16; ++s) {
            const int dch = 4 * s + 2 * lh;
            aReg[s].x = zp[(size_t)dch * HW];
            aReg[s].y = zp[(size_t)(dch + 1) * HW];
        }

        float minv[8];
        int   mini[8];
#pragma unroll
        for (int r = 0; r < 8; ++r) { minv[r] = 3.4e38f; mini[r] = 0; }

#pragma unroll 1
        for (int c = 0; c < KCODE / CHUNK; ++c) {
            const int cbase = c * CHUNK;
            __syncthreads();  // previous chunk fully consumed
            // stage -2*e, swizzled: (code,d) -> code*64 + (d ^ (code&63))
            const float* ec = emb + (size_t)cbase * DCH;
            for (int e2 = tid; e2 < CHUNK * DCH; e2 += 256) {
                const int cl = e2 >> 6, dd = e2 & 63;
                ldsB[(cl << 6) | (dd ^ (cl & 63))] = -2.0f * ec[e2];
            }
            if (tid < CHUNK) {  // ||e||^2 from global (L2-hot)
                const float* er = emb + (size_t)(cbase + tid) * DCH;
                float s = 0.0f;
#pragma unroll
                for (int dd = 0; dd < DCH; dd += 4) {
                    v4f v = *(const v4f*)(er + dd);
                    s += v.x * v.x + v.y * v.y + v.z * v.z + v.w * v.w;
                }
                ldsE[tid] = s;
            }
            __syncthreads();

#pragma unroll 1
            for (int kt = 0; kt < CHUNK / 16; ++kt) {
                const int codeL = kt * 16 + col;
                const float en = ldsE[codeL];
                v8f acc = {en, en, en, en, en, en, en, en};
                const float* bp = &ldsB[codeL << 6];
                const int sw = codeL & 63;
#pragma unroll
                for (int s = 0; s < 16; ++s) {
                    const int kk = 4 * s + 2 * lh;   // B layout: K = kk, kk+1
                    v2f bv;
                    bv.x = bp[kk ^ sw];
                    bv.y = bp[(kk + 1) ^ sw];
                    acc = __builtin_amdgcn_wmma_f32_16x16x4_f32(
                        false, aReg[s], false, bv, (short)0, acc, false, false);
                }
                const int gcode = cbase + codeL;
#pragma unroll
                for (int r = 0; r < 8; ++r) {
                    if (acc[r] < minv[r]) { minv[r] = acc[r]; mini[r] = gcode; }
                }
            }
        }

        // ---- cross-lane argmin within each 16-lane half (rows r / r+8),
        // first-index tie-break to match jnp.argmin.
#pragma unroll
        for (int r = 0; r < 8; ++r) {
            float mv = minv[r]; int mi = mini[r];
#pragma unroll
            for (int off = 1; off < 16; off <<= 1) {
                const float ov = __shfl_xor(mv, off, 32);
                const int   oi = __shfl_xor(mi, off, 32);
                if (ov < mv || (ov == mv && oi < mi)) { mv = ov; mi = oi; }
            }
            minv[r] = mv; mini[r] = mi;
        }
        if (col == 0) {  // lane 0 -> rows 0..7, lane 16 -> rows 8..15
            const int base = t16 + lh * 8;
#pragma unroll
            for (int r = 0; r < 8; ++r) {
                const int n = base + r;
                wsIdx[n]   = mini[r];
                outIdxF[n] = (float)mini[r];
                atomicAdd(&hist[mini[r]], 1u);
            }
        }
    }
}

// ---------------------------------------------------------------------------
// Kernel 3: one-hot min_encodings. 268MB of NT stores (won't fit 192MB L2).
// Region base is only 4B-aligned, so scalar b32 NT stores, each instruction
// unit-stride across lanes for perfect coalescing.
__global__ __launch_bounds__(256) void vq_onehot(
    const int* __restrict__ wsIdx, float* __restrict__ outEnc)
{
    const int n   = blockIdx.x;
    const int idx = wsIdx[n];
    float* row = outEnc + (size_t)n * KCODE;
    const int k = threadIdx.x;
#pragma unroll
    for (int j = 0; j < 4; ++j) {
        const int kk = k + j * 256;
        __builtin_nontemporal_store((kk == idx) ? 1.0f : 0.0f, row + kk);
    }
}

// ---------------------------------------------------------------------------
// Kernel 4: gather z_q (== z_q_st forward value, buggy flat layout) + loss
// partials. Deterministic: one partial per block, summed in fixed order later.
__global__ __launch_bounds__(256) void vq_gather_loss(
    const float* __restrict__ z, const float* __restrict__ emb,
    const int* __restrict__ wsIdx, float* __restrict__ outZq,
    float* __restrict__ partials)
{
    __shared__ float red[256];
    const int i4 = blockIdx.x * 256 + threadIdx.x;
    const size_t i = (size_t)i4 * 4;
    const int n = (int)(i >> 6);
    const int d = (int)(i & 63);
    const int idx = wsIdx[n];
    const v4f e  = *(const v4f*)(emb + (size_t)idx * DCH + d);
    const v4f zv = *(const v4f*)(z + i);
    *(v4f*)(outZq + i) = e;
    const v4f df = e - zv;
    red[threadIdx.x] = df.x * df.x + df.y * df.y + df.z * df.z + df.w * df.w;
    __syncthreads();
    for (int o = 128; o > 0; o >>= 1) {
        if (threadIdx.x < o) red[threadIdx.x] += red[threadIdx.x + o];
        __syncthreads();
    }
    if (threadIdx.x == 0) partials[blockIdx.x] = red[0];
}

// ---------------------------------------------------------------------------
// Kernel 5: finalize perplexity + loss (fixed-order reductions).
__global__ __launch_bounds__(1024) void vq_finalize(
    const unsigned* __restrict__ hist, const float* __restrict__ partials,
    float* __restrict__ outPerp, float* __restrict__ outLoss)
{
    __shared__ float red[1024];
    const int t = threadIdx.x;
    red[t] = partials[t] + partials[t + 1024] + partials[t + 2048] + partials[t + 3072];
    __syncthreads();
    for (int o = 512; o > 0; o >>= 1) {
        if (t < o) red[t] += red[t + o];
        __syncthreads();
    }
    const float lossSum = red[0];
    __syncthreads();
    const float p = (float)hist[t] * (1.0f / (float)NPTS);
    red[t] = p * logf(p + 1e-10f);
    __syncthreads();
    for (int o = 512; o > 0; o >>= 1) {
        if (t < o) red[t] += red[t + o];
        __syncthreads();
    }
    if (t == 0) {
        outPerp[0] = expf(-red[0]);
        // loss = mean((zq-z)^2) + BETA*mean((z-zq)^2) = 1.5 * mean(diff^2)
        outLoss[0] = 1.5f * lossSum * (1.0f / (float)NTOT);
    }
}

// ---------------------------------------------------------------------------
extern "C" void kernel_launch(void* const* d_in, const int* in_sizes, int n_in,
                              void* d_out, int out_size, void* d_ws, size_t ws_size,
                              hipStream_t stream) {
    const float* z   = (const float*)d_in[0];
    const float* emb = (const float*)d_in[1];
    float* out = (float*)d_out;

    // workspace layout
    int*      wsIdx    = (int*)d_ws;                                   // 65536*4
    unsigned* hist     = (unsigned*)((char*)d_ws + (size_t)NPTS * 4);  // 1024*4
    float*    partials = (float*)((char*)d_ws + (size_t)NPTS * 4 + 4096); // 4096*4

    // output layout (flat, return order)
    float* outZq   = out;
    float* outPerp = out + 4194304;
    float* outIdx  = out + 4194305;
    float* outEnc  = out + 4259841;
    float* outLoss = out + 71368705;

    vq_init<<<1, 1024, 0, stream>>>(hist);
    vq_argmin<<<GRID_ARGMIN, 256, 0, stream>>>(z, emb, wsIdx, outIdx, hist);
    vq_onehot<<<NPTS, 256, 0, stream>>>(wsIdx, outEnc);
    vq_gather_loss<<<NTOT / 1024, 256, 0, stream>>>(z, emb, wsIdx, outZq, partials);
    vq_finalize<<<1, 1024, 0, stream>>>(hist, partials, outPerp, outLoss);
}